// Pixel_spareformer_54073638257175
// MI455X (gfx1250) — compile-verified
//
#include <hip/hip_runtime.h>
#include <hip/hip_bf16.h>

#define N_NODES 100000
#define N_EDGES 1200000
#define DIN 128
#define DOUT 64
#define EPSV 1e-5f

typedef __attribute__((ext_vector_type(2))) float v2f;
typedef __attribute__((ext_vector_type(8))) float v8f;

__device__ __forceinline__ void atomicMaxF32(float* addr, float val) {
    // ordered-float trick: works for mixed signs with init = -inf
    if (val >= 0.0f) atomicMax((int*)addr, __float_as_int(val));
    else             atomicMin((unsigned int*)addr, __float_as_uint(val));
}

// ---------------- init: zero output + BN accumulators ----------------
__global__ void k_init(float* __restrict__ out, float* __restrict__ bnsum,
                       float* __restrict__ bnsq) {
    long long i = (long long)blockIdx.x * blockDim.x + threadIdx.x;
    const long long total = (long long)N_NODES * 2 * DOUT;
    for (long long j = i; j < total; j += (long long)gridDim.x * blockDim.x)
        out[j] = 0.0f;
    if (blockIdx.x == 0 && threadIdx.x < 2 * DOUT) {
        bnsum[threadIdx.x] = 0.0f;
        bnsq[threadIdx.x]  = 0.0f;
    }
}

__global__ void k_seginit(float* __restrict__ segmax, float* __restrict__ denom) {
    int i = blockIdx.x * blockDim.x + threadIdx.x;
    if (i < N_NODES) {
        segmax[i] = -__builtin_huge_valf();
        denom[i]  = 0.0f;
    }
}

// ---------------- GEMM: out[N,64] = x[N,128] @ W[128,64] + b ----------------
// block = 128 threads = 4 waves; wave w owns column tile w (16 cols).
// 16-row x tile staged in LDS (stride 132 to avoid bank conflicts).
// WMMA f32 16x16x4, K looped 128/4 = 32 chained MMAs.
__global__ __launch_bounds__(128)
void k_gemm(const float* __restrict__ x, const float* __restrict__ W,
            const float* __restrict__ b, float* __restrict__ out) {
    __shared__ float lds[16 * 132];
    const int m0  = blockIdx.x * 16;
    const int tid = threadIdx.x;
    for (int i = tid; i < 16 * 128; i += 128) {
        int r = i >> 7, c = i & 127;
        lds[r * 132 + c] = x[(long long)(m0 + r) * DIN + c];
    }
    __syncthreads();

    const int lane = tid & 31;
    const int wave = tid >> 5;      // column tile 0..3
    const int half = lane >> 4;     // 0: K pair {0,1}; 1: K pair {2,3}
    const int lm   = lane & 15;
    const int col  = wave * 16 + lm;

    v8f acc;
    const float bias = b[col];
#pragma unroll
    for (int r = 0; r < 8; ++r) acc[r] = bias;

#pragma unroll
    for (int k0 = 0; k0 < DIN; k0 += 4) {
        const int kr = k0 + half * 2;
        v2f A, B;
        // A 16x4 f32: lanes0-15 K={k0,k0+1}, lanes16-31 K={k0+2,k0+3}, M = lane%16
        A.x = lds[lm * 132 + kr];
        A.y = lds[lm * 132 + kr + 1];
        // B 4x16 f32: row (N) striped across lanes within a VGPR
        B.x = W[kr * DOUT + col];
        B.y = W[(kr + 1) * DOUT + col];
        acc = __builtin_amdgcn_wmma_f32_16x16x4_f32(
            false, A, false, B, (short)0, acc, false, false);
    }

    // C/D: VGPR r holds M = r (lanes0-15) / M = r+8 (lanes16-31), N = lane%16
    const int rowBase = m0 + half * 8;
#pragma unroll
    for (int r = 0; r < 8; ++r)
        out[(long long)(rowBase + r) * DOUT + col] = acc[r];
}

// ---------------- LayerNorm over last dim (64) : wave per row ----------------
__global__ void k_layernorm(float* __restrict__ q) {
    const int gw   = (blockIdx.x * blockDim.x + threadIdx.x) >> 5;
    const int nw   = (gridDim.x * blockDim.x) >> 5;
    const int lane = threadIdx.x & 31;
    for (int row = gw; row < N_NODES; row += nw) {
        float2 xv = ((const float2*)(q + (long long)row * DOUT))[lane];
        float s  = xv.x + xv.y;
        float ss = xv.x * xv.x + xv.y * xv.y;
        for (int m = 16; m; m >>= 1) {
            s  += __shfl_xor(s, m, 32);
            ss += __shfl_xor(ss, m, 32);
        }
        const float mean = s * (1.0f / DOUT);
        const float var  = ss * (1.0f / DOUT) - mean * mean;
        const float inv  = rsqrtf(var + EPSV);
        xv.x = (xv.x - mean) * inv;
        xv.y = (xv.y - mean) * inv;
        ((float2*)(q + (long long)row * DOUT))[lane] = xv;
    }
}

// ---------------- scores + segment max : wave per edge ----------------
__global__ void k_scores(const float* __restrict__ q, const int* __restrict__ src,
                         const int* __restrict__ dst, float* __restrict__ scores,
                         float* __restrict__ segmax) {
    const int gw   = (blockIdx.x * blockDim.x + threadIdx.x) >> 5;
    const int nw   = (gridDim.x * blockDim.x) >> 5;
    const int lane = threadIdx.x & 31;
    for (int e = gw; e < N_EDGES; e += nw) {
        const int s = src[e], d = dst[e];
        float2 a = ((const float2*)(q + (long long)s * DOUT))[lane];
        float2 b = ((const float2*)(q + (long long)d * DOUT))[lane];
        float p = a.x * b.x + a.y * b.y;
        for (int m = 16; m; m >>= 1) p += __shfl_xor(p, m, 32);
        p *= (1.0f / DOUT);
        if (lane == 0) {
            scores[e] = p;
            atomicMaxF32(&segmax[s], p);
        }
    }
}

// ---------------- exp + segment denom : thread per edge ----------------
__global__ void k_expdenom(float* __restrict__ scores, const int* __restrict__ src,
                           const float* __restrict__ segmax, float* __restrict__ denom) {
    const int i = blockIdx.x * blockDim.x + threadIdx.x;
    if (i >= N_EDGES) return;
    const int s = src[i];
    const float e = __expf(scores[i] - segmax[s]);
    scores[i] = e;
    atomicAdd(&denom[s], e);
}

// ---------------- weighted scatter-add : wave per edge ----------------
__global__ void k_agg(const float* __restrict__ v, const float* __restrict__ scores,
                      const float* __restrict__ denom, const int* __restrict__ src,
                      const int* __restrict__ dst, float* __restrict__ out,
                      const int colOff) {
    const int gw   = (blockIdx.x * blockDim.x + threadIdx.x) >> 5;
    const int nw   = (gridDim.x * blockDim.x) >> 5;
    const int lane = threadIdx.x & 31;
    for (int e = gw; e < N_EDGES; e += nw) {
        const int s = src[e], d = dst[e];
        const float att = scores[e] / denom[s];
        float2 vv = ((const float2*)(v + (long long)d * DOUT))[lane];
        float* o = out + (long long)s * (2 * DOUT) + colOff + lane * 2;
        atomicAdd(o,     vv.x * att);
        atomicAdd(o + 1, vv.y * att);
    }
}

// ---------------- BN stats: per-feature sum/sumsq over N rows ----------------
__global__ void k_bnstats(const float* __restrict__ out, float* __restrict__ bnsum,
                          float* __restrict__ bnsq) {
    __shared__ float ls[256], lq[256];
    const int f = threadIdx.x & 127;              // feature
    const int rofs = threadIdx.x >> 7;            // 0/1 — 2 rows per iter
    float s = 0.0f, q = 0.0f;
    for (int r = blockIdx.x * 2 + rofs; r < N_NODES; r += gridDim.x * 2) {
        const float v = out[(long long)r * 128 + f];
        s += v;
        q += v * v;
    }
    ls[threadIdx.x] = s;
    lq[threadIdx.x] = q;
    __syncthreads();
    if (threadIdx.x < 128) {
        atomicAdd(&bnsum[f], ls[threadIdx.x] + ls[threadIdx.x + 128]);
        atomicAdd(&bnsq[f],  lq[threadIdx.x] + lq[threadIdx.x + 128]);
    }
}

__global__ void k_bnfinal(const float* __restrict__ bnsum, const float* __restrict__ bnsq,
                          const float* __restrict__ gamma, const float* __restrict__ beta,
                          float* __restrict__ scale, float* __restrict__ shift) {
    const int f = threadIdx.x;
    if (f >= 128) return;
    const float mean = bnsum[f] * (1.0f / N_NODES);
    const float var  = bnsq[f] * (1.0f / N_NODES) - mean * mean;
    const float sc   = gamma[f] * rsqrtf(var + EPSV);
    scale[f] = sc;
    shift[f] = beta[f] - mean * sc;
}

__global__ void k_bnapply(float* __restrict__ out, const float* __restrict__ scale,
                          const float* __restrict__ shift) {
    const long long i = (long long)blockIdx.x * blockDim.x + threadIdx.x;
    if (i >= (long long)N_NODES * 128) return;
    const int f = (int)(i & 127);
    float y = out[i] * scale[f] + shift[f];
    out[i] = (y > 0.0f) ? y : 0.01f * y;
}

extern "C" void kernel_launch(void* const* d_in, const int* in_sizes, int n_in,
                              void* d_out, int out_size, void* d_ws, size_t ws_size,
                              hipStream_t stream) {
    const float* x      = (const float*)d_in[0];
    const int*   rowsrc = (const int*)d_in[1];
    const int*   rowdst = (const int*)d_in[2];
    const int*   colsrc = (const int*)d_in[3];
    const int*   coldst = (const int*)d_in[4];
    const float* Wrv    = (const float*)d_in[5];
    const float* brv    = (const float*)d_in[6];
    const float* Wcv    = (const float*)d_in[7];
    const float* bcv    = (const float*)d_in[8];
    const float* Wrq    = (const float*)d_in[9];
    const float* brq    = (const float*)d_in[10];
    const float* Wcq    = (const float*)d_in[11];
    const float* bcq    = (const float*)d_in[12];
    const float* gamma  = (const float*)d_in[13];
    const float* beta   = (const float*)d_in[14];
    float* out = (float*)d_out;

    // workspace carve-up (floats)
    float* ws      = (float*)d_ws;
    float* q       = ws;
    float* v       = q + (size_t)N_NODES * DOUT;
    float* sc      = v + (size_t)N_NODES * DOUT;
    float* segmax  = sc + (size_t)N_EDGES;
    float* denom   = segmax + N_NODES;
    float* bnsum   = denom + N_NODES;
    float* bnsq    = bnsum + 2 * DOUT;
    float* bnscale = bnsq + 2 * DOUT;
    float* bnshift = bnscale + 2 * DOUT;

    k_init<<<1024, 256, 0, stream>>>(out, bnsum, bnsq);

    for (int branch = 0; branch < 2; ++branch) {
        const int*   src = branch ? colsrc : rowsrc;
        const int*   dst = branch ? coldst : rowdst;
        const float* Wv  = branch ? Wcv : Wrv;
        const float* bv  = branch ? bcv : brv;
        const float* Wq  = branch ? Wcq : Wrq;
        const float* bq  = branch ? bcq : brq;

        k_gemm<<<N_NODES / 16, 128, 0, stream>>>(x, Wv, bv, v);
        k_gemm<<<N_NODES / 16, 128, 0, stream>>>(x, Wq, bq, q);
        k_layernorm<<<1024, 256, 0, stream>>>(q);
        k_seginit<<<(N_NODES + 255) / 256, 256, 0, stream>>>(segmax, denom);
        k_scores<<<4096, 256, 0, stream>>>(q, src, dst, sc, segmax);
        k_expdenom<<<(N_EDGES + 255) / 256, 256, 0, stream>>>(sc, src, segmax, denom);
        k_agg<<<4096, 256, 0, stream>>>(v, sc, denom, src, dst, out, branch * DOUT);
    }

    k_bnstats<<<1024, 256, 0, stream>>>(out, bnsum, bnsq);
    k_bnfinal<<<1, 128, 0, stream>>>(bnsum, bnsq, gamma, beta, bnscale, bnshift);
    const long long tot = (long long)N_NODES * 128;
    k_bnapply<<<(int)((tot + 255) / 256), 256, 0, stream>>>(out, bnscale, bnshift);
}